// GNN_11141145165946
// MI455X (gfx1250) — compile-verified
//
#include <hip/hip_runtime.h>

typedef float v2f __attribute__((ext_vector_type(2)));
typedef float v8f __attribute__((ext_vector_type(8)));

#define HID 64

// ---------------- degree / norm ----------------
__global__ void k_init_deg(float* __restrict__ deg, int n) {
    int i = blockIdx.x * blockDim.x + threadIdx.x;
    if (i < n) deg[i] = 1.0f;  // self-loop contributes 1
}

__global__ void k_deg_edges(const int* __restrict__ dst, float* __restrict__ deg, int e) {
    int i = blockIdx.x * blockDim.x + threadIdx.x;
    if (i < e) atomicAdd(&deg[dst[i]], 1.0f);
}

__global__ void k_rsqrt(float* __restrict__ deg, int n) {
    int i = blockIdx.x * blockDim.x + threadIdx.x;
    if (i < n) {
        float d = deg[i];
        deg[i] = (d > 0.0f) ? rsqrtf(d) : 0.0f;
    }
}

// ------------- self-loop + bias init: agg[i][f] = h[i][f]*dinv[i]^2 + b[f] -------------
__global__ void k_self_bias(const float* __restrict__ h, const float* __restrict__ dinv,
                            const float* __restrict__ b, float* __restrict__ agg, int n) {
    int idx = blockIdx.x * blockDim.x + threadIdx.x;
    if (idx >= n * HID) return;
    int node = idx >> 6;
    int f    = idx & 63;
    float s = dinv[node];
    agg[idx] = h[idx] * s * s + b[f];
}

// ------------- edge scatter: one wave32 per edge, 2 feats/lane -------------
__global__ void k_scatter(const float* __restrict__ h, const float* __restrict__ dinv,
                          const int* __restrict__ src, const int* __restrict__ dst,
                          float* __restrict__ agg, int e) {
    int eid = blockIdx.x * (blockDim.x >> 5) + (threadIdx.x >> 5);
    if (eid >= e) return;
    int lane = threadIdx.x & 31;
    int s = src[eid];
    int d = dst[eid];
    float norm = dinv[s] * dinv[d];
    float2 hv = ((const float2*)(h + (size_t)s * HID))[lane];
    float* out = agg + (size_t)d * HID + lane * 2;
    atomicAdd(out,     hv.x * norm);
    atomicAdd(out + 1, hv.y * norm);
}

// ------------- WMMA GEMM: C[N,64] = act(A[N,K]) @ B[K,64] (+bias) -------------
// grid.x = N/16 (N multiple of 16), block = 128 (4 waves); wave w -> 16x16 tile cols [16w,16w+16)
__global__ void k_gemm_wmma(const float* __restrict__ A, const float* __restrict__ B,
                            float* __restrict__ C, int K, int ksh,
                            const float* __restrict__ bias, int relu_in) {
    __shared__ float As[16 * 130];  // max K=128, row stride K+2 (conflict-free, 8B aligned)
    const int lds  = K + 2;
    const int row0 = blockIdx.x << 4;

    // cooperative coalesced load of contiguous 16xK A tile, fused input ReLU
    const float* Ab = A + (size_t)row0 * K;
    const int total = 16 << ksh;
    for (int i = threadIdx.x; i < total; i += blockDim.x) {
        float v = Ab[i];
        if (relu_in) v = fmaxf(v, 0.0f);
        As[(i >> ksh) * lds + (i & (K - 1))] = v;
    }
    __syncthreads();

    const int wave  = threadIdx.x >> 5;
    const int lane  = threadIdx.x & 31;
    const int m16   = lane & 15;
    const int n     = m16 + (wave << 4);       // output column
    const int khalf = (lane >> 4) << 1;        // 0 for lanes 0-15, 2 for lanes 16-31

    v8f c = {};
    for (int k0 = 0; k0 < K; k0 += 4) {
        const int kk = k0 + khalf;
        // A frag: lane holds A[m16][kk], A[m16][kk+1]
        float2 af = *(const float2*)&As[m16 * lds + kk];
        v2f a; a[0] = af.x; a[1] = af.y;
        // B frag: lane holds B[kk][n], B[kk+1][n]  (lanes 0-15: K rows k0,k0+1; 16-31: k0+2,k0+3)
        v2f b; b[0] = B[kk * HID + n]; b[1] = B[(kk + 1) * HID + n];
        c = __builtin_amdgcn_wmma_f32_16x16x4_f32(false, a, false, b, (short)0, c,
                                                  false, false);
    }

    // C/D layout: VGPR r -> row r (+8 for upper half-wave), col n
    float bv = bias ? bias[n] : 0.0f;
    const int mb = (lane >> 4) << 3;
    float* Crow = C + (size_t)row0 * HID;
#pragma unroll
    for (int r = 0; r < 8; ++r) {
        Crow[(size_t)(mb + r) * HID + n] = c[r] + bv;
    }
}

extern "C" void kernel_launch(void* const* d_in, const int* in_sizes, int n_in,
                              void* d_out, int out_size, void* d_ws, size_t ws_size,
                              hipStream_t stream) {
    const float* x   = (const float*)d_in[0];
    const int*   ei  = (const int*)d_in[1];
    const float* W1  = (const float*)d_in[2];
    const float* b1  = (const float*)d_in[3];
    const float* W2  = (const float*)d_in[4];
    const float* b2  = (const float*)d_in[5];
    const float* Wfc = (const float*)d_in[6];
    const float* bfc = (const float*)d_in[7];
    float* out = (float*)d_out;

    const int N = in_sizes[0] / 128;   // 100000, divisible by 16
    const int E = in_sizes[1] / 2;     // 1600000
    const int* src = ei;
    const int* dst = ei + E;

    // workspace: dinv[N] | h[N*64] | agg[N*64]
    char* ws = (char*)d_ws;
    size_t off0 = ((size_t)N * sizeof(float) + 255) & ~(size_t)255;
    float* dinv = (float*)ws;
    float* h    = (float*)(ws + off0);
    float* agg  = (float*)(ws + off0 + (size_t)N * HID * sizeof(float));

    const int T = 256;
    const int nb_nodes = (N + T - 1) / T;
    const int nb_edges = (E + T - 1) / T;
    const int nb_feat  = (int)(((size_t)N * HID + T - 1) / T);
    const int nb_scat  = (E + 7) / 8;       // 8 waves (edges) per 256-thread block
    const int nb_gemm  = N / 16;

    // degree -> dinv
    k_init_deg <<<nb_nodes, T, 0, stream>>>(dinv, N);
    k_deg_edges<<<nb_edges, T, 0, stream>>>(dst, dinv, E);
    k_rsqrt    <<<nb_nodes, T, 0, stream>>>(dinv, N);

    // layer 1: h = x @ W1 ; agg = selfloop + b1 ; scatter edges
    k_gemm_wmma<<<nb_gemm, 128, 0, stream>>>(x, W1, h, 128, 7, nullptr, 0);
    k_self_bias<<<nb_feat, T, 0, stream>>>(h, dinv, b1, agg, N);
    k_scatter  <<<nb_scat, 256, 0, stream>>>(h, dinv, src, dst, agg, E);

    // layer 2: h = relu(agg) @ W2 (relu fused) ; agg = selfloop + b2 ; scatter
    k_gemm_wmma<<<nb_gemm, 128, 0, stream>>>(agg, W2, h, 64, 6, nullptr, 1);
    k_self_bias<<<nb_feat, T, 0, stream>>>(h, dinv, b2, agg, N);
    k_scatter  <<<nb_scat, 256, 0, stream>>>(h, dinv, src, dst, agg, E);

    // final FC: out = relu(agg) @ Wfc + bfc (relu + bias fused)
    k_gemm_wmma<<<nb_gemm, 128, 0, stream>>>(agg, Wfc, out, 64, 6, bfc, 1);
}